// LogSig_v2_5111011082836
// MI455X (gfx1250) — compile-verified
//
#include <hip/hip_runtime.h>

typedef __attribute__((ext_vector_type(2))) float v2f;
typedef __attribute__((ext_vector_type(8))) float v8f;

#define BATCH   256
#define NTIME   301
#define DCH     96
#define NSEG    50
#define OUTF    (96 + (96 * 95) / 2)   // 4656 floats per (b,s)
#define LROW    100                    // padded LDS row stride (floats)
#define WAVES   4

// Output element (i,j), i<j, lives at: 96 + i*95 - i(i-1)/2 + (j-i-1)
//                                    = triu_rowbase(i) + j
__host__ __device__ constexpr int triu_rowbase(int i) {
    return 96 + i * 95 - (i * (i - 1)) / 2 - i - 1;
}

__global__ __launch_bounds__(WAVES * 32)
void logsig_wmma_kernel(const float* __restrict__ inp, float* __restrict__ out)
{
    __shared__ float lds[WAVES][7 * LROW];

    const int lane = threadIdx.x & 31;
    const int wave = threadIdx.x >> 5;
    const int gw   = blockIdx.x * WAVES + wave;      // 0 .. 12799 (grid exact)
    const int b    = gw / NSEG;
    const int s    = gw - b * NSEG;

    float* L = lds[wave];
    const float* src = inp + ((size_t)b * NTIME + (size_t)(6 * s)) * DCH;

    // Segment rows 6s..6s+6 are contiguous: 672 floats. Coalesced b32 loads.
    #pragma unroll
    for (int t = 0; t < 7; ++t) {
        #pragma unroll
        for (int c0 = 0; c0 < 96; c0 += 32)
            L[t * LROW + c0 + lane] = src[t * 96 + c0 + lane];
    }
    __syncthreads();   // cross-lane LDS visibility (per-wave slices, one barrier)

    const int   half = lane >> 4;      // 0: K=0,1 (h0) / K=4,5 (h1);  1: K=2,3 / K=6,7
    const int   lcol = lane & 15;      // A role: row m.  B role: col n.
    const int   k0   = 2 * half;       // 0 or 2
    const float msk  = (half == 0) ? 1.0f : 0.0f;  // K=6,7 are zero padding
    const int   r4   = (half == 0) ? 4 : 0;        // clamped rows (avoid OOB LDS)
    const int   r5   = (half == 0) ? 5 : 0;
    const int   r6   = (half == 0) ? 6 : 0;

    // Per 16-wide tile of the D=96 axis: fragments for xprev and dx, 2 K-halves.
    v2f xp[6][2], dxv[6][2];
    #pragma unroll
    for (int tt = 0; tt < 6; ++tt) {
        const int   c  = 16 * tt + lcol;
        const float p0 = L[c];                       // seg[0][c]
        const float a0 = L[k0 * LROW + c];           // seg[k0]
        const float a1 = L[(k0 + 1) * LROW + c];     // seg[k0+1]
        const float a2 = L[(k0 + 2) * LROW + c];     // seg[k0+2]
        v2f t;
        t[0] = a0 - p0;          t[1] = a1 - p0;          xp[tt][0]  = t;  // xprev k0,k0+1
        t[0] = a1 - a0;          t[1] = a2 - a1;          dxv[tt][0] = t;  // dx    k0,k0+1
        const float q4 = L[r4 * LROW + c];
        const float q5 = L[r5 * LROW + c];
        const float q6 = L[r6 * LROW + c];
        t[0] = msk * (q4 - p0);  t[1] = msk * (q5 - p0);  xp[tt][1]  = t;  // xprev 4,5 (6,7 = 0)
        t[0] = msk * (q5 - q4);  t[1] = msk * (q6 - q5);  dxv[tt][1] = t;  // dx    4,5 (6,7 = 0)
    }

    float* ob = out + (size_t)gw * OUTF;

    // lvl1 = seg[6] - seg[0]
    #pragma unroll
    for (int c0 = 0; c0 < 96; c0 += 32)
        ob[c0 + lane] = L[6 * LROW + c0 + lane] - L[c0 + lane];

    // A = 0.5*(M - M^T); use (xp ⊗ dx)^T = dx ⊗ xp to avoid transposes.
    #pragma unroll
    for (int I = 0; I < 6; ++I) {
        // Per-(I,r) output row pointer: i = 16I + r + 8*half is fixed across J,
        // so the triu row base is a lane-select between two compile-time
        // constants; the 16J column step folds into the store immediate.
        float* rowp[8];
        #pragma unroll
        for (int r = 0; r < 8; ++r) {
            const int i0 = 16 * I + r;                     // compile-time
            const int Ca = triu_rowbase(i0);               // half == 0
            const int Cb = triu_rowbase(i0 + 8);           // half == 1
            rowp[r] = ob + ((half ? Cb : Ca) + lcol);
        }

        // ---- Diagonal tile (J == I): strict-upper filter needed ----
        {
            v8f acc1 = {};
            acc1 = __builtin_amdgcn_wmma_f32_16x16x4_f32(
                       false, xp[I][0], false, dxv[I][0], (short)0, acc1, false, false);
            acc1 = __builtin_amdgcn_wmma_f32_16x16x4_f32(
                       false, xp[I][1], false, dxv[I][1], (short)0, acc1, false, false);
            v8f acc2 = {};
            acc2 = __builtin_amdgcn_wmma_f32_16x16x4_f32(
                       false, dxv[I][0], false, xp[I][0], (short)0, acc2, false, false);
            acc2 = __builtin_amdgcn_wmma_f32_16x16x4_f32(
                       false, dxv[I][1], false, xp[I][1], (short)0, acc2, false, false);
            #pragma unroll
            for (int r = 0; r < 8; ++r) {
                if (r + 8 * half < lcol)   // m < n within the diagonal tile
                    rowp[r][16 * I] = 0.5f * (acc1[r] - acc2[r]);
            }
        }

        // ---- Off-diagonal tiles (J > I): every element valid, no predicate ----
        #pragma unroll
        for (int J = I + 1; J < 6; ++J) {
            v8f acc1 = {};   // M[16I:,16J:] = sum_k xprev[k][i] * dx[k][j]
            acc1 = __builtin_amdgcn_wmma_f32_16x16x4_f32(
                       false, xp[I][0], false, dxv[J][0], (short)0, acc1, false, false);
            acc1 = __builtin_amdgcn_wmma_f32_16x16x4_f32(
                       false, xp[I][1], false, dxv[J][1], (short)0, acc1, false, false);
            v8f acc2 = {};   // M^T tile = sum_k dx[k][i] * xprev[k][j]
            acc2 = __builtin_amdgcn_wmma_f32_16x16x4_f32(
                       false, dxv[I][0], false, xp[J][0], (short)0, acc2, false, false);
            acc2 = __builtin_amdgcn_wmma_f32_16x16x4_f32(
                       false, dxv[I][1], false, xp[J][1], (short)0, acc2, false, false);
            #pragma unroll
            for (int r = 0; r < 8; ++r)
                rowp[r][16 * J] = 0.5f * (acc1[r] - acc2[r]);
        }
    }
}

extern "C" void kernel_launch(void* const* d_in, const int* in_sizes, int n_in,
                              void* d_out, int out_size, void* d_ws, size_t ws_size,
                              hipStream_t stream)
{
    (void)in_sizes; (void)n_in; (void)out_size; (void)d_ws; (void)ws_size;
    const float* inp = (const float*)d_in[0];   // (256, 301, 96) f32
    float*       out = (float*)d_out;           // (256, 50, 4656) f32

    const int total_waves = BATCH * NSEG;       // 12800
    dim3 grid(total_waves / WAVES);             // 3200
    dim3 block(WAVES * 32);                     // 128
    hipLaunchKernelGGL(logsig_wmma_kernel, grid, block, 0, stream, inp, out);
}